// SPVCNN_MAE_22978075034245
// MI455X (gfx1250) — compile-verified
//
#include <hip/hip_runtime.h>

// SPVCNN scale-block for MI455X (gfx1250, wave32, WMMA).
// Memory-bound workload -> fp32 WMMA (v_wmma_f32_16x16x4_f32) keeps reference
// precision at zero cost; fuse gather+3 GEMMs+scatter in one kernel.

#define N_PREV  250000
#define N_FULL  500000
#define NVOX    65536
#define HCAP    (1u << 19)
#define HMASK   (HCAP - 1u)
#define NT_PREV (N_PREV / 16)   // 15625 exact
#define NT_FULL (N_FULL / 16)   // 31250 exact
#define FUSED_WAVES 4
#define PP_WAVES 8

typedef __attribute__((ext_vector_type(2))) float v2f;
typedef __attribute__((ext_vector_type(8))) float v8f;

__device__ __forceinline__ float lrelu(float x) { return x > 0.f ? x : 0.1f * x; }
__device__ __forceinline__ v8f vz8() { v8f z = {0.f,0.f,0.f,0.f,0.f,0.f,0.f,0.f}; return z; }

// ---- fp32 WMMA tile GEMM: [16 x K] (LDS, row-major stride K) x [K x NT*16]
// (LDS, transposed weights Wt[n*LDB + k]) accumulated into NT 16x16 f32 tiles.
// A frag (16x4 f32): lane L -> row m=L&15, k = kbase + 2*(L>>4) + {0,1}
// B frag (4x16 f32): lane L -> col n=L&15, k = kbase + 2*(L>>4) + {0,1}
template<int K, int NT, int LDB>
__device__ __forceinline__ void wave_gemm(const float* As, const float* Wt,
                                          int kpos0, v8f* acc, int lane) {
  const int ln = lane & 15;
  const int hk = (lane >> 4) << 1;
#pragma unroll
  for (int kk = 0; kk < K / 4; ++kk) {
    v2f a = *(const v2f*)(As + ln * K + kk * 4 + hk);
#pragma unroll
    for (int nt = 0; nt < NT; ++nt) {
      v2f b = *(const v2f*)(Wt + (nt * 16 + ln) * LDB + kpos0 + kk * 4 + hk);
      acc[nt] = __builtin_amdgcn_wmma_f32_16x16x4_f32(
          false, a, false, b, (short)0, acc[nt], false, false);
    }
  }
}

__device__ __forceinline__ int voxel_key(const int* coors, int i) {
  int b = coors[i * 4 + 0];
  int x = coors[i * 4 + 1] >> 1;
  int y = coors[i * 4 + 2] >> 1;
  int z = coors[i * 4 + 3] >> 1;
  return ((b * 501 + x) * 501 + y) * 31 + z;
}
__device__ __forceinline__ unsigned khash(int key) {
  return ((unsigned)key * 2654435761u) & HMASK;
}

// ---------------- init fills ----------------
__global__ void k_fill_f32(float* p, float v, size_t n) {
  size_t i = (size_t)blockIdx.x * blockDim.x + threadIdx.x;
  size_t st = (size_t)gridDim.x * blockDim.x;
  for (; i < n; i += st) p[i] = v;
}
__global__ void k_fill_i32(int* p, int v, size_t n) {
  size_t i = (size_t)blockIdx.x * blockDim.x + threadIdx.x;
  size_t st = (size_t)gridDim.x * blockDim.x;
  for (; i < n; i += st) p[i] = v;
}

// ---------------- hash build ----------------
__global__ void k_insert(const int* __restrict__ coors, int* __restrict__ hkeys) {
  int i = blockIdx.x * blockDim.x + threadIdx.x;
  if (i >= N_PREV) return;
  int key = voxel_key(coors, i);
  unsigned slot = khash(key);
  for (;;) {
    int prev = atomicCAS(&hkeys[slot], -1, key);
    if (prev == -1 || prev == key) break;
    slot = (slot + 1) & HMASK;
  }
}

__global__ void k_assign(const int* __restrict__ hkeys, int* __restrict__ hids,
                         int* __restrict__ counters) {
  int s = blockIdx.x * blockDim.x + threadIdx.x;
  if (s < (int)HCAP && hkeys[s] != -1) hids[s] = atomicAdd(&counters[0], 1);
}

// ---------------- pooling (scatter-sum + counts) ----------------
__global__ void k_pool(const float* __restrict__ feat, const int* __restrict__ coors,
                       const int* __restrict__ hkeys, const int* __restrict__ hids,
                       int* __restrict__ inv, float* __restrict__ cnt,
                       float* __restrict__ pooled) {
  int i = blockIdx.x * blockDim.x + threadIdx.x;
  if (i >= N_PREV) return;
  int key = voxel_key(coors, i);
  unsigned slot = khash(key);
  while (hkeys[slot] != key) slot = (slot + 1) & HMASK;
  int seg = hids[slot];
  inv[i] = seg;
  atomicAdd(&cnt[seg], 1.0f);
  const float* fr = feat + (size_t)i * 64;
  float* pr = pooled + (size_t)seg * 64;
#pragma unroll 8
  for (int c = 0; c < 64; ++c) atomicAdd(&pr[c], fr[c]);
}

// ---------------- PPmodel layer 1: pooled-mean @ W1 (64->32), lrelu, BN stats ----------------
__global__ void __launch_bounds__(256)
k_pp1(const float* __restrict__ pooled, const float* __restrict__ cnt,
      const float* __restrict__ W1, const float* __restrict__ b1,
      const int* __restrict__ counters, float* __restrict__ h1,
      float* __restrict__ gstats) {
  __shared__ __align__(16) float sWt[32 * 64];
  __shared__ float sb[32];
  __shared__ float sst[64];
  __shared__ __align__(16) float sA[PP_WAVES][16 * 64];
  const int tid = threadIdx.x;
  for (int t = tid; t < 2048; t += 256) { int k = t >> 5, n = t & 31; sWt[n * 64 + k] = W1[t]; }
  if (tid < 32) sb[tid] = b1[tid];
  if (tid < 64) sst[tid] = 0.f;
  __syncthreads();
  const int wv = tid >> 5, lane = tid & 31;
  const int tile = blockIdx.x * PP_WAVES + wv;
  if (tile < NT_PREV) {
    const int seg0 = tile * 16;
    float* As = sA[wv];
    for (int m = 0; m < 16; ++m) {
      float rc = 1.0f / fmaxf(cnt[seg0 + m], 1.0f);
      v2f p = *(const v2f*)(pooled + (size_t)(seg0 + m) * 64 + lane * 2);
      v2f o; o[0] = p[0] * rc; o[1] = p[1] * rc;
      *(v2f*)(As + m * 64 + lane * 2) = o;
    }
    __builtin_amdgcn_wave_barrier();
    v8f acc[2]; acc[0] = vz8(); acc[1] = vz8();
    wave_gemm<64, 2, 64>(As, sWt, 0, acc, lane);
    const int nu = counters[0];
    const int ln = lane & 15, hf = lane >> 4;
#pragma unroll
    for (int nt = 0; nt < 2; ++nt) {
      int n = nt * 16 + ln;
      float bb = sb[n];
      float s = 0.f, q = 0.f;
#pragma unroll
      for (int i = 0; i < 8; ++i) {
        int seg = seg0 + i + hf * 8;
        float x = lrelu(acc[nt][i] + bb);
        h1[(size_t)seg * 32 + n] = x;
        if (seg < nu) { s += x; q += x * x; }
      }
      atomicAdd(&sst[n], s);
      atomicAdd(&sst[32 + n], q);
    }
  }
  __syncthreads();
  if (tid < 64) atomicAdd(&gstats[tid], sst[tid]);
}

// ---------------- BN params: scale = g*rsqrt(var+eps), shift = be - mu*scale ----------------
__global__ void k_bnparams(const float* __restrict__ gstats, const float* __restrict__ g,
                           const float* __restrict__ be, const int* __restrict__ counters,
                           float* __restrict__ bnp, int C) {
  int c = threadIdx.x;
  if (c >= C) return;
  int ni = counters[0]; if (ni < 1) ni = 1;
  float n = (float)ni;
  float mu = gstats[c] / n;
  float var = fmaxf(gstats[C + c] / n - mu * mu, 0.f);
  float sc = g[c] * rsqrtf(var + 1e-5f);
  bnp[c] = sc;
  bnp[C + c] = be[c] - mu * sc;
}

// ---------------- PPmodel layer 2: BN(h1) @ W2 (32->32), lrelu, BN stats ----------------
__global__ void __launch_bounds__(256)
k_pp2(const float* __restrict__ h1, const float* __restrict__ bnp,
      const float* __restrict__ W2, const float* __restrict__ b2,
      const int* __restrict__ counters, float* __restrict__ h2,
      float* __restrict__ gstats) {
  __shared__ __align__(16) float sWt[32 * 32];
  __shared__ float sb[32];
  __shared__ float sst[64];
  __shared__ __align__(16) float sA[PP_WAVES][16 * 32];
  const int tid = threadIdx.x;
  for (int t = tid; t < 1024; t += 256) { int k = t >> 5, n = t & 31; sWt[n * 32 + k] = W2[t]; }
  if (tid < 32) sb[tid] = b2[tid];
  if (tid < 64) sst[tid] = 0.f;
  __syncthreads();
  const int wv = tid >> 5, lane = tid & 31;
  const int tile = blockIdx.x * PP_WAVES + wv;
  if (tile < NT_PREV) {
    const int seg0 = tile * 16;
    float* As = sA[wv];
    const float scl = bnp[lane], sh = bnp[32 + lane];
    for (int m = 0; m < 16; ++m) {
      float x = h1[(size_t)(seg0 + m) * 32 + lane];
      As[m * 32 + lane] = x * scl + sh;
    }
    __builtin_amdgcn_wave_barrier();
    v8f acc[2]; acc[0] = vz8(); acc[1] = vz8();
    wave_gemm<32, 2, 32>(As, sWt, 0, acc, lane);
    const int nu = counters[0];
    const int ln = lane & 15, hf = lane >> 4;
#pragma unroll
    for (int nt = 0; nt < 2; ++nt) {
      int n = nt * 16 + ln;
      float bb = sb[n];
      float s = 0.f, q = 0.f;
#pragma unroll
      for (int i = 0; i < 8; ++i) {
        int seg = seg0 + i + hf * 8;
        float x = lrelu(acc[nt][i] + bb);
        h2[(size_t)seg * 32 + n] = x;
        if (seg < nu) { s += x; q += x * x; }
      }
      atomicAdd(&sst[n], s);
      atomicAdd(&sst[32 + n], q);
    }
  }
  __syncthreads();
  if (tid < 64) atomicAdd(&gstats[tid], sst[tid]);
}

// ---------------- PPmodel layer 3: BN(h2) @ W3 (32->64), lrelu ----------------
__global__ void __launch_bounds__(256)
k_pp3(const float* __restrict__ h2, const float* __restrict__ bnp,
      const float* __restrict__ W3, const float* __restrict__ b3v,
      float* __restrict__ h3) {
  __shared__ __align__(16) float sWt[64 * 32];
  __shared__ float sb[64];
  __shared__ __align__(16) float sA[PP_WAVES][16 * 32];
  const int tid = threadIdx.x;
  for (int t = tid; t < 2048; t += 256) { int k = t >> 6, n = t & 63; sWt[n * 32 + k] = W3[t]; }
  if (tid < 64) sb[tid] = b3v[tid];
  __syncthreads();
  const int wv = tid >> 5, lane = tid & 31;
  const int tile = blockIdx.x * PP_WAVES + wv;
  if (tile >= NT_PREV) return;
  const int seg0 = tile * 16;
  float* As = sA[wv];
  const float scl = bnp[lane], sh = bnp[32 + lane];
  for (int m = 0; m < 16; ++m) {
    float x = h2[(size_t)(seg0 + m) * 32 + lane];
    As[m * 32 + lane] = x * scl + sh;
  }
  __builtin_amdgcn_wave_barrier();
  v8f acc[4]; acc[0] = vz8(); acc[1] = vz8(); acc[2] = vz8(); acc[3] = vz8();
  wave_gemm<32, 4, 32>(As, sWt, 0, acc, lane);
  const int ln = lane & 15, hf = lane >> 4;
#pragma unroll
  for (int nt = 0; nt < 4; ++nt) {
    int n = nt * 16 + ln;
    float bb = sb[n];
#pragma unroll
    for (int i = 0; i < 8; ++i)
      h3[(size_t)(seg0 + i + hf * 8) * 64 + n] = lrelu(acc[nt][i] + bb);
  }
}

// ---------------- fused per-point pipeline over N_FULL ----------------
// per 16-pt tile: gather feat/h3 rows -> LDS; identity = lrelu(feat@W_in+b_in);
// t = lrelu(identity@Wo1[0:64] + h@Wo1[64:128] + bo1); u = t@Wo2 + bo2;
// atomic scatter u into voxel accumulators.
// dynamic LDS: WtI 4096 | WtO1 8192 | WtO2 4096 | biases 192 | per-wave 2112
#define FUSED_SMEM_FLOATS (16576 + FUSED_WAVES * 2112)
__global__ void __launch_bounds__(32 * FUSED_WAVES)
k_fused(const float* __restrict__ feat, const int* __restrict__ cif,
        const int* __restrict__ scv, const int* __restrict__ inv,
        const float* __restrict__ h3,
        const float* __restrict__ W_in, const float* __restrict__ b_in,
        const float* __restrict__ Wo1, const float* __restrict__ bo1,
        const float* __restrict__ Wo2, const float* __restrict__ bo2,
        float* __restrict__ voxAcc, int* __restrict__ voxCnt) {
  extern __shared__ __align__(16) float sm[];
  float* WtI  = sm;           // [64][64] transposed W_in
  float* WtO1 = sm + 4096;    // [64][128] transposed Wo1
  float* WtO2 = sm + 12288;   // [64][64] transposed Wo2
  float* bIs  = sm + 16384;
  float* bO1s = sm + 16448;
  float* bO2s = sm + 16512;
  const int tid = threadIdx.x;
  const int bdim = 32 * FUSED_WAVES;
  for (int t = tid; t < 4096; t += bdim) {
    int k = t >> 6, n = t & 63;
    WtI[n * 64 + k] = W_in[t];
    WtO2[n * 64 + k] = Wo2[t];
  }
  for (int t = tid; t < 8192; t += bdim) {
    int k = t >> 6, n = t & 63;
    WtO1[n * 128 + k] = Wo1[t];
  }
  for (int t = tid; t < 64; t += bdim) { bIs[t] = b_in[t]; bO1s[t] = bo1[t]; bO2s[t] = bo2[t]; }
  __syncthreads();

  const int wv = tid >> 5, lane = tid & 31;
  const int tile = blockIdx.x * FUSED_WAVES + wv;
  if (tile >= NT_FULL) return;

  float* wbase = sm + 16576 + wv * 2112;
  float* Ast = wbase;          // 16x64 staging (feat -> identity)
  float* Bst = wbase + 1024;   // 16x64 staging (h rows -> lrelu(t))
  int* meta = (int*)(wbase + 2048);  // [0:16) row idx, [16:32) seg, [32:48) voxel
  const int ln = lane & 15, hf = lane >> 4;

  if (lane < 16) {
    int p = tile * 16 + lane;
    int r = cif[p];
    meta[lane] = r;
    meta[16 + lane] = inv[r];
    meta[32 + lane] = scv[p];
  }
  __builtin_amdgcn_wave_barrier();

  for (int m = 0; m < 16; ++m) {
    int r = meta[m];
    *(v2f*)(Ast + m * 64 + lane * 2) = *(const v2f*)(feat + (size_t)r * 64 + lane * 2);
    int s = meta[16 + m];
    *(v2f*)(Bst + m * 64 + lane * 2) = *(const v2f*)(h3 + (size_t)s * 64 + lane * 2);
  }
  __builtin_amdgcn_wave_barrier();

  // GEMM1: identity = lrelu(feat @ W_in + b_in)
  v8f acc0[4]; acc0[0] = vz8(); acc0[1] = vz8(); acc0[2] = vz8(); acc0[3] = vz8();
  wave_gemm<64, 4, 64>(Ast, WtI, 0, acc0, lane);
  __builtin_amdgcn_wave_barrier();
#pragma unroll
  for (int nt = 0; nt < 4; ++nt) {
    float bb = bIs[nt * 16 + ln];
#pragma unroll
    for (int i = 0; i < 8; ++i)
      Ast[(i + hf * 8) * 64 + nt * 16 + ln] = lrelu(acc0[nt][i] + bb);
  }
  __builtin_amdgcn_wave_barrier();

  // GEMM2: t = identity @ Wo1[0:64,:] + h @ Wo1[64:128,:]
  v8f acc1[4]; acc1[0] = vz8(); acc1[1] = vz8(); acc1[2] = vz8(); acc1[3] = vz8();
  wave_gemm<64, 4, 128>(Ast, WtO1, 0, acc1, lane);
  wave_gemm<64, 4, 128>(Bst, WtO1, 64, acc1, lane);
  __builtin_amdgcn_wave_barrier();
#pragma unroll
  for (int nt = 0; nt < 4; ++nt) {
    float bb = bO1s[nt * 16 + ln];
#pragma unroll
    for (int i = 0; i < 8; ++i)
      Bst[(i + hf * 8) * 64 + nt * 16 + ln] = lrelu(acc1[nt][i] + bb);
  }
  __builtin_amdgcn_wave_barrier();

  // GEMM3: u = lrelu(t) @ Wo2 + bo2 ; scatter into voxels
  v8f acc2[4]; acc2[0] = vz8(); acc2[1] = vz8(); acc2[2] = vz8(); acc2[3] = vz8();
  wave_gemm<64, 4, 64>(Bst, WtO2, 0, acc2, lane);
#pragma unroll
  for (int nt = 0; nt < 4; ++nt) {
    int n = nt * 16 + ln;
    float bb = bO2s[n];
#pragma unroll
    for (int i = 0; i < 8; ++i) {
      int v = meta[32 + i + hf * 8];
      atomicAdd(voxAcc + (size_t)v * 64 + n, acc2[nt][i] + bb);
    }
  }
  if (lane < 16) atomicAdd(&voxCnt[meta[32 + lane]], 1);
}

// ---------------- finalize: voxel mean ----------------
__global__ void k_finalize(const float* __restrict__ voxAcc, const int* __restrict__ voxCnt,
                           float* __restrict__ out, size_t n) {
  size_t i = (size_t)blockIdx.x * blockDim.x + threadIdx.x;
  size_t st = (size_t)gridDim.x * blockDim.x;
  for (; i < n; i += st) {
    int v = (int)(i >> 6);
    out[i] = voxAcc[i] / fmaxf((float)voxCnt[v], 1.0f);
  }
}

extern "C" void kernel_launch(void* const* d_in, const int* in_sizes, int n_in,
                              void* d_out, int out_size, void* d_ws, size_t ws_size,
                              hipStream_t stream) {
  (void)in_sizes; (void)n_in; (void)out_size; (void)ws_size;
  const float* feat = (const float*)d_in[0];
  const int*   coors = (const int*)d_in[1];
  const int*   cif   = (const int*)d_in[2];
  const int*   scv   = (const int*)d_in[3];
  const float* W_in = (const float*)d_in[4];
  const float* b_in = (const float*)d_in[5];
  const float* W1 = (const float*)d_in[6];
  const float* b1 = (const float*)d_in[7];
  const float* g1 = (const float*)d_in[8];
  const float* be1 = (const float*)d_in[9];
  const float* W2 = (const float*)d_in[10];
  const float* b2 = (const float*)d_in[11];
  const float* g2 = (const float*)d_in[12];
  const float* be2 = (const float*)d_in[13];
  const float* W3 = (const float*)d_in[14];
  const float* b3 = (const float*)d_in[15];
  const float* Wo1 = (const float*)d_in[16];
  const float* bo1 = (const float*)d_in[17];
  const float* Wo2 = (const float*)d_in[18];
  const float* bo2 = (const float*)d_in[19];
  float* out = (float*)d_out;

  char* base = (char*)d_ws;
  size_t off = 0;
  auto alloc = [&](size_t bytes) -> void* {
    void* p = base + off;
    off = (off + bytes + 255) & ~(size_t)255;
    return p;
  };
  int*   hkeys   = (int*)alloc(HCAP * 4);
  int*   hids    = (int*)alloc(HCAP * 4);
  int*   counters= (int*)alloc(256);
  int*   inv     = (int*)alloc((size_t)N_PREV * 4);
  float* cnt     = (float*)alloc((size_t)N_PREV * 4);
  float* pooled  = (float*)alloc((size_t)N_PREV * 64 * 4);
  float* h1      = (float*)alloc((size_t)N_PREV * 32 * 4);
  float* h2      = (float*)alloc((size_t)N_PREV * 32 * 4);
  float* h3      = (float*)alloc((size_t)N_PREV * 64 * 4);
  float* st1     = (float*)alloc(64 * 4);
  float* st2     = (float*)alloc(64 * 4);
  float* bnp1    = (float*)alloc(64 * 4);
  float* bnp2    = (float*)alloc(64 * 4);
  float* voxAcc  = (float*)alloc((size_t)NVOX * 64 * 4);
  int*   voxCnt  = (int*)alloc((size_t)NVOX * 4);

  // init
  k_fill_i32<<<2048, 256, 0, stream>>>(hkeys, -1, (size_t)HCAP);
  k_fill_i32<<<1, 64, 0, stream>>>(counters, 0, 64);
  k_fill_f32<<<512, 256, 0, stream>>>(cnt, 0.f, (size_t)N_PREV);
  k_fill_f32<<<4096, 256, 0, stream>>>(pooled, 0.f, (size_t)N_PREV * 64);
  k_fill_f32<<<1, 64, 0, stream>>>(st1, 0.f, 64);
  k_fill_f32<<<1, 64, 0, stream>>>(st2, 0.f, 64);
  k_fill_f32<<<4096, 256, 0, stream>>>(voxAcc, 0.f, (size_t)NVOX * 64);
  k_fill_i32<<<256, 256, 0, stream>>>(voxCnt, 0, (size_t)NVOX);

  // voxel unique + pooling
  k_insert<<<(N_PREV + 255) / 256, 256, 0, stream>>>(coors, hkeys);
  k_assign<<<HCAP / 256, 256, 0, stream>>>(hkeys, hids, counters);
  k_pool<<<(N_PREV + 255) / 256, 256, 0, stream>>>(feat, coors, hkeys, hids, inv, cnt, pooled);

  // PPmodel (WMMA) + masked BN
  const int ppBlocks = (NT_PREV + PP_WAVES - 1) / PP_WAVES;
  k_pp1<<<ppBlocks, 256, 0, stream>>>(pooled, cnt, W1, b1, counters, h1, st1);
  k_bnparams<<<1, 64, 0, stream>>>(st1, g1, be1, counters, bnp1, 32);
  k_pp2<<<ppBlocks, 256, 0, stream>>>(h1, bnp1, W2, b2, counters, h2, st2);
  k_bnparams<<<1, 64, 0, stream>>>(st2, g2, be2, counters, bnp2, 32);
  k_pp3<<<ppBlocks, 256, 0, stream>>>(h2, bnp2, W3, b3, h3);

  // fused gather + 3 GEMMs + voxel scatter
  const int fusedBlocks = (NT_FULL + FUSED_WAVES - 1) / FUSED_WAVES;
  k_fused<<<fusedBlocks, 32 * FUSED_WAVES, FUSED_SMEM_FLOATS * sizeof(float), stream>>>(
      feat, cif, scv, inv, h3, W_in, b_in, Wo1, bo1, Wo2, bo2, voxAcc, voxCnt);

  // voxel mean
  k_finalize<<<4096, 256, 0, stream>>>(voxAcc, voxCnt, out, (size_t)NVOX * 64);
}